// Lipschitz_loss_43542378447380
// MI455X (gfx1250) — compile-verified
//
#include <hip/hip_runtime.h>

#define D_IN   37632
#define E_DIM  512
#define NM     128      // rows of x  (B*2)
#define NN     64       // rows of y  (B)
#define COS_EPS 1e-8f
#define LIPC    0.5f

typedef __attribute__((ext_vector_type(2))) float v2f;
typedef __attribute__((ext_vector_type(8))) float v8f;

// ---------------------------------------------------------------- zero fill
__global__ void zero_kernel(float* __restrict__ g, int n) {
    int i = blockIdx.x * blockDim.x + threadIdx.x;
    if (i < n) g[i] = 0.f;
}

// ------------------------------------------- per-batch row reductions over D
// yn2[b]   = |inp[b,0]|^2
// xl2[2b+l]= |inp[b,1+l]|^2
// diffp2   = |inp[b,1+l]-inp[b,0]|^2   (direct, as reference does)
// maskf    = any(inp[b,1+l] != inp[b,0]) ? 1 : 0
__global__ void row_reduce_kernel(const float* __restrict__ inp,
                                  float* __restrict__ yn2, float* __restrict__ xl2,
                                  float* __restrict__ diffp2, float* __restrict__ maskf)
{
    __shared__ float    rs[5][256];
    __shared__ unsigned ms[2][256];
    const int b = blockIdx.x, tid = threadIdx.x;
    const float4* y  = (const float4*)(inp + (size_t)b * 3 * D_IN);
    const float4* x1 = (const float4*)(inp + (size_t)b * 3 * D_IN + D_IN);
    const float4* x2 = (const float4*)(inp + (size_t)b * 3 * D_IN + 2 * D_IN);
    float sy = 0.f, s1 = 0.f, s2 = 0.f, d1 = 0.f, d2 = 0.f;
    unsigned m1 = 0u, m2 = 0u;
    for (int i = tid; i < D_IN / 4; i += 256) {
        float4 yv = y[i], v1 = x1[i], v2 = x2[i];
        sy += yv.x*yv.x + yv.y*yv.y + yv.z*yv.z + yv.w*yv.w;
        s1 += v1.x*v1.x + v1.y*v1.y + v1.z*v1.z + v1.w*v1.w;
        s2 += v2.x*v2.x + v2.y*v2.y + v2.z*v2.z + v2.w*v2.w;
        float ax = v1.x - yv.x, ay = v1.y - yv.y, az = v1.z - yv.z, aw = v1.w - yv.w;
        d1 += ax*ax + ay*ay + az*az + aw*aw;
        float bx = v2.x - yv.x, by = v2.y - yv.y, bz = v2.z - yv.z, bw = v2.w - yv.w;
        d2 += bx*bx + by*by + bz*bz + bw*bw;
        m1 |= (unsigned)((v1.x != yv.x) | (v1.y != yv.y) | (v1.z != yv.z) | (v1.w != yv.w));
        m2 |= (unsigned)((v2.x != yv.x) | (v2.y != yv.y) | (v2.z != yv.z) | (v2.w != yv.w));
    }
    rs[0][tid]=sy; rs[1][tid]=s1; rs[2][tid]=s2; rs[3][tid]=d1; rs[4][tid]=d2;
    ms[0][tid]=m1; ms[1][tid]=m2;
    __syncthreads();
    for (int s = 128; s > 0; s >>= 1) {
        if (tid < s) {
            rs[0][tid]+=rs[0][tid+s]; rs[1][tid]+=rs[1][tid+s]; rs[2][tid]+=rs[2][tid+s];
            rs[3][tid]+=rs[3][tid+s]; rs[4][tid]+=rs[4][tid+s];
            ms[0][tid]|=ms[0][tid+s]; ms[1][tid]|=ms[1][tid+s];
        }
        __syncthreads();
    }
    if (tid == 0) {
        yn2[b] = rs[0][0];
        xl2[2*b]   = rs[1][0];  xl2[2*b+1]   = rs[2][0];
        diffp2[2*b]= rs[3][0];  diffp2[2*b+1]= rs[4][0];
        maskf[2*b]   = ms[0][0] ? 1.f : 0.f;
        maskf[2*b+1] = ms[1][0] ? 1.f : 0.f;
    }
}

// ------------------------------- normalize `out` rows, record norms of result
__global__ void normalize_kernel(const float* __restrict__ outp,
                                 float* __restrict__ hn, float* __restrict__ lnv,
                                 float* __restrict__ na, float* __restrict__ nb)
{
    __shared__ float red[128];
    const int j = blockIdx.x;            // 0..191 = (b, li)
    const int b = j / 3, li = j % 3;
    const float* row = outp + (size_t)j * E_DIM;
    const int tid = threadIdx.x;         // 128 threads
    float ss = 0.f;
    for (int e = tid; e < E_DIM; e += 128) { float v = row[e]; ss += v * v; }
    red[tid] = ss; __syncthreads();
    for (int s = 64; s > 0; s >>= 1) { if (tid < s) red[tid] += red[tid + s]; __syncthreads(); }
    const float norm = sqrtf(red[0]);
    __syncthreads();
    float* dst = (li == 0) ? (hn + (size_t)b * E_DIM)
                           : (lnv + (size_t)(b * 2 + li - 1) * E_DIM);
    float nss = 0.f;
    for (int e = tid; e < E_DIM; e += 128) {
        float v = row[e] / norm;
        dst[e] = v;
        nss += v * v;
    }
    red[tid] = nss; __syncthreads();
    for (int s = 64; s > 0; s >>= 1) { if (tid < s) red[tid] += red[tid + s]; __syncthreads(); }
    if (tid == 0) {
        float nn = sqrtf(red[0]);
        if (li == 0) na[b] = nn; else nb[b * 2 + li - 1] = nn;
    }
}

// --------------------------------------------- fp32 WMMA GEMM, LDS-staged
// MODE 0: A row m -> inp[(m>>1), 1+(m&1), :], B row n -> inp[n, 0, :], split-K atomics
// MODE 1: A row m -> lnv[m,:],               B row n -> hn[n,:],       direct store
template<int MODE>
__global__ void wmma_gemm_kernel(const float* __restrict__ A,
                                 const float* __restrict__ Bm,
                                 float* __restrict__ G,
                                 int chunks)
{
    constexpr int KC = 32;
    constexpr int LDP = KC + 4;   // 36 floats/row: 16B-aligned rows, bank-conflict-free
    __shared__ __align__(16) float Xs[16][LDP];
    __shared__ __align__(16) float Ys[16][LDP];

    const int lane = threadIdx.x;     // 0..31, one wave per block
    const int mt = blockIdx.x;
    const int nt = blockIdx.y;
    const int ks = blockIdx.z;

    // LDS loaders: thread -> (row lr, half-row lc), 16 contiguous floats each
    const int lr = lane >> 1;
    const int lc = (lane & 1) * 16;

    const int am = mt * 16 + lr;
    const int bn = nt * 16 + lr;
    const float* arow;
    const float* brow;
    if (MODE == 0) {
        arow = A  + (size_t)((am >> 1) * 3 + 1 + (am & 1)) * D_IN;
        brow = Bm + (size_t)bn * 3 * D_IN;
    } else {
        arow = A  + (size_t)am * E_DIM;
        brow = Bm + (size_t)bn * E_DIM;
    }

    const int half = lane >> 4;
    const int r    = lane & 15;
    const int k2   = 2 * half;

    v8f acc = {0.f,0.f,0.f,0.f,0.f,0.f,0.f,0.f};

    int k0 = ks * chunks * KC;
    for (int c = 0; c < chunks; ++c, k0 += KC) {
        const float4* ag = (const float4*)(arow + k0 + lc);
        const float4* bg = (const float4*)(brow + k0 + lc);
        float4 a0 = ag[0], a1 = ag[1], a2 = ag[2], a3 = ag[3];
        float4 b0 = bg[0], b1 = bg[1], b2 = bg[2], b3 = bg[3];
        __syncthreads();   // WAR vs previous iteration's LDS reads
        *(float4*)&Xs[lr][lc +  0] = a0; *(float4*)&Xs[lr][lc +  4] = a1;
        *(float4*)&Xs[lr][lc +  8] = a2; *(float4*)&Xs[lr][lc + 12] = a3;
        *(float4*)&Ys[lr][lc +  0] = b0; *(float4*)&Ys[lr][lc +  4] = b1;
        *(float4*)&Ys[lr][lc +  8] = b2; *(float4*)&Ys[lr][lc + 12] = b3;
        __syncthreads();
        #pragma unroll
        for (int kk = 0; kk < KC; kk += 4) {
            v2f a, bb;
            a.x  = Xs[r][kk + k2];  a.y  = Xs[r][kk + k2 + 1];
            bb.x = Ys[r][kk + k2];  bb.y = Ys[r][kk + k2 + 1];
            acc = __builtin_amdgcn_wmma_f32_16x16x4_f32(
                false, a, false, bb, (short)0, acc, false, false);
        }
    }

    const int gn = nt * 16 + r;
    #pragma unroll
    for (int v = 0; v < 8; ++v) {
        int gm = mt * 16 + half * 8 + v;
        if (MODE == 0) atomicAdd(&G[gm * NN + gn], acc[v]);
        else           G[gm * NN + gn] = acc[v];
    }
}

// ------------------------------------------------------------- final scalar
__global__ void finalize_kernel(const int* __restrict__ labels,
                                const float* __restrict__ G1, const float* __restrict__ G2,
                                const float* __restrict__ xl2, const float* __restrict__ yn2,
                                const float* __restrict__ diffp2, const float* __restrict__ maskf,
                                const float* __restrict__ na, const float* __restrict__ nb,
                                float* __restrict__ outv)
{
    __shared__ float wsh[64];
    __shared__ int   lab[64];
    __shared__ float red[256];
    const int tid = threadIdx.x;
    if (tid < 64) lab[tid] = labels[tid];
    __syncthreads();
    if (tid < 64) {
        int c = 0, ml = lab[tid];
        for (int j = 0; j < 64; ++j) c += (lab[j] == ml) ? 1 : 0;
        wsh[tid] = (float)(c > 8 ? 8 : c);   // count of validness slots > 0
    }
    __syncthreads();
    float acc = 0.f;
    // positive: cos_p[b,l] = G2[m, b] / max(na*nb, eps)
    for (int m = tid; m < NM; m += 256) {
        int b = m >> 1;
        float cosp = G2[m * NN + b] / fmaxf(na[b] * nb[m], COS_EPS);
        float pen  = fmaxf((1.f - cosp) / sqrtf(diffp2[m]) - LIPC, 0.f);
        acc += maskf[m] * pen;
    }
    // negative: sum over n != b of hinge((1-cos_n)/diff_n - LIP) * w[n]
    for (int idx = tid; idx < NM * NN; idx += 256) {
        int m = idx >> 6, n = idx & 63;
        int b = m >> 1;
        if (n != b) {
            float sq   = xl2[m] + yn2[n] - 2.f * G1[idx];
            float diff = sqrtf(fmaxf(sq, 0.f));
            float cosn = G2[idx] / fmaxf(nb[m] * na[n], COS_EPS);
            float pen  = fmaxf((1.f - cosn) / diff - LIPC, 0.f) * wsh[n];
            acc -= maskf[m] * pen;
        }
    }
    red[tid] = acc; __syncthreads();
    for (int s = 128; s > 0; s >>= 1) { if (tid < s) red[tid] += red[tid + s]; __syncthreads(); }
    if (tid == 0) outv[0] = red[0] * (1.f / 128.f);
}

extern "C" void kernel_launch(void* const* d_in, const int* in_sizes, int n_in,
                              void* d_out, int out_size, void* d_ws, size_t ws_size,
                              hipStream_t stream)
{
    const float* inp    = (const float*)d_in[0];   // (64,3,37632)
    const float* outp   = (const float*)d_in[1];   // (64,3,512)
    const int*   labels = (const int*)d_in[2];     // (64,)
    // d_in[3] (memory_bank_HR) and d_in[4] (validness) are dead in the forward value.

    float* ws     = (float*)d_ws;
    float* G1     = ws;                    // 128*64
    float* G2     = ws + 8192;             // 128*64
    float* xl2    = ws + 16384;            // 128
    float* yn2    = ws + 16512;            // 64
    float* diffp2 = ws + 16576;            // 128
    float* maskf  = ws + 16704;            // 128
    float* na     = ws + 16832;            // 64
    float* nb     = ws + 16896;            // 128
    float* hn     = ws + 17024;            // 64*512
    float* lnv    = ws + 17024 + 64*512;   // 128*512

    zero_kernel<<<32, 256, 0, stream>>>(G1, 8192);
    row_reduce_kernel<<<64, 256, 0, stream>>>(inp, yn2, xl2, diffp2, maskf);
    normalize_kernel<<<192, 128, 0, stream>>>(outp, hn, lnv, na, nb);
    // K = 37632 = 56 slices * 21 chunks * KC(32); split-K via f32 atomics
    wmma_gemm_kernel<0><<<dim3(8, 4, 56), 32, 0, stream>>>(inp, inp, G1, 21);
    // K = 512 = 16 chunks * 32; single slice, direct store
    wmma_gemm_kernel<1><<<dim3(8, 4, 1), 32, 0, stream>>>(lnv, hn, G2, 16);
    finalize_kernel<<<1, 256, 0, stream>>>(labels, G1, G2, xl2, yn2, diffp2, maskf,
                                           na, nb, (float*)d_out);
}